// GNNRewriteDiscriminator_29025388986755
// MI455X (gfx1250) — compile-verified
//
#include <hip/hip_runtime.h>

// ---------------------------------------------------------------------------
// GNN rewrite discriminator: twin GCN encoder + MLP head, CDNA5 (gfx1250).
// Dense transforms use V_WMMA_F32_16X16X4_F32 (fp32, no precision loss).
// Edge aggregation is L2-resident atomic gather/scatter (bandwidth bound).
// ---------------------------------------------------------------------------

typedef __attribute__((ext_vector_type(2))) float v2f;
typedef __attribute__((ext_vector_type(8))) float v8f;

#define HCH   32          // hidden channels
#define INCH  16          // in channels

static __device__ __forceinline__ v8f wmma4(v2f a, v2f b, v8f c) {
  // D = A(16x4,f32) * B(4x16,f32) + C(16x16,f32)
  return __builtin_amdgcn_wmma_f32_16x16x4_f32(
      /*neg_a=*/false, a, /*neg_b=*/false, b,
      /*c_mod=*/(short)0, c, /*reuse_a=*/false, /*reuse_b=*/false);
}

// ------------------------------ elementwise --------------------------------

__global__ void k_set(float* p, float v, int n) {
  int i = blockIdx.x * blockDim.x + threadIdx.x;
  if (i < n) p[i] = v;
}

__global__ void k_degree(const int* __restrict__ dst, float* deg, int E) {
  int e = blockIdx.x * blockDim.x + threadIdx.x;
  if (e < E) atomicAdd(&deg[dst[e]], 1.0f);
}

__global__ void k_rsqrt(float* d, int n) {
  int i = blockIdx.x * blockDim.x + threadIdx.x;
  if (i < n) d[i] = rsqrtf(d[i]);
}

// -------------------------- WMMA dense transforms --------------------------
// One wave handles a 16-row x 32-col output tile; K stepped by 4.
// A layout (16x4 f32): lanes 0-15 -> M, VGPR0 holds K=k0 (lo half)/k0+2 (hi),
// VGPR1 holds K=k0+1 / k0+3.  B (4x16): row%2 -> VGPR, row/2 -> lane half.
// C/D (16x16): VGPR j holds M=j (lanes0-15) / M=j+8 (lanes16-31), N = lane%16.
// Grids are arranged so every wave is fully active (no divergence; EXEC=~0).

// out[N,32] = emb[tok[i],:16] @ W[16,32]
__global__ void k_embed_gemm(const int* __restrict__ tok,
                             const float* __restrict__ emb,
                             const float* __restrict__ W,
                             float* __restrict__ out) {
  int wave = blockIdx.x * (blockDim.x >> 5) + (threadIdx.x >> 5);
  int lane = threadIdx.x & 31;
  int half = lane >> 4;
  int m    = lane & 15;
  int row0 = wave * 16;

  const float* arow = emb + (long long)tok[row0 + m] * INCH;

  v8f c0 = {0.f,0.f,0.f,0.f,0.f,0.f,0.f,0.f};
  v8f c1 = c0;
#pragma unroll
  for (int k0 = 0; k0 < INCH; k0 += 4) {
    int ka = k0 + 2 * half;
    v2f a;  a[0]  = arow[ka];            a[1]  = arow[ka + 1];
    v2f b0; b0[0] = W[ka * HCH + m];     b0[1] = W[(ka + 1) * HCH + m];
    v2f b1; b1[0] = W[ka * HCH + 16 + m];b1[1] = W[(ka + 1) * HCH + 16 + m];
    c0 = wmma4(a, b0, c0);
    c1 = wmma4(a, b1, c1);
  }
#pragma unroll
  for (int j = 0; j < 8; ++j) {
    int r = row0 + j + 8 * half;
    out[(long long)r * HCH + m]      = c0[j];
    out[(long long)r * HCH + 16 + m] = c1[j];
  }
}

// out[rows,32] = A[rows,K] @ W[K,32]   (K multiple of 4; rows multiple of 32)
__global__ void k_gemm32(const float* __restrict__ A,
                         const float* __restrict__ W,
                         float* __restrict__ out, int K) {
  int wave = blockIdx.x * (blockDim.x >> 5) + (threadIdx.x >> 5);
  int lane = threadIdx.x & 31;
  int half = lane >> 4;
  int m    = lane & 15;
  int row0 = wave * 16;

  const float* arow = A + (long long)(row0 + m) * K;

  v8f c0 = {0.f,0.f,0.f,0.f,0.f,0.f,0.f,0.f};
  v8f c1 = c0;
  for (int k0 = 0; k0 < K; k0 += 4) {
    int ka = k0 + 2 * half;
    v2f a;  a[0]  = arow[ka];            a[1]  = arow[ka + 1];
    v2f b0; b0[0] = W[ka * HCH + m];     b0[1] = W[(ka + 1) * HCH + m];
    v2f b1; b1[0] = W[ka * HCH + 16 + m];b1[1] = W[(ka + 1) * HCH + 16 + m];
    c0 = wmma4(a, b0, c0);
    c1 = wmma4(a, b1, c1);
  }
#pragma unroll
  for (int j = 0; j < 8; ++j) {
    int r = row0 + j + 8 * half;
    out[(long long)r * HCH + m]      = c0[j];
    out[(long long)r * HCH + 16 + m] = c1[j];
  }
}

// ---------------------------- edge aggregation -----------------------------
// 8 lanes/edge, float4 gather, 4 atomic f32 adds each (L2-resident).

__global__ void k_scatter(const float* __restrict__ xw,
                          const int* __restrict__ src,
                          const int* __restrict__ dst,
                          const float* __restrict__ dinv,
                          float* __restrict__ out, int E) {
  int gid  = blockIdx.x * blockDim.x + threadIdx.x;
  int e    = gid >> 3;
  if (e >= E) return;
  int part = gid & 7;
  int s = src[e], d = dst[e];
  float nrm = dinv[s] * dinv[d];
  float4 v = ((const float4*)(xw + (long long)s * HCH))[part];
  float* o = out + (long long)d * HCH + part * 4;
  atomicAdd(o + 0, v.x * nrm);
  atomicAdd(o + 1, v.y * nrm);
  atomicAdd(o + 2, v.z * nrm);
  atomicAdd(o + 3, v.w * nrm);
}

// out[i,h] += xw[i,h]*dinv[i]^2 + b[h]; optional ReLU (second conv)
__global__ void k_selfbias(const float* __restrict__ xw,
                           const float* __restrict__ dinv,
                           const float* __restrict__ b,
                           float* __restrict__ out, int N, int relu) {
  int gid = blockIdx.x * blockDim.x + threadIdx.x;
  if (gid >= N * HCH) return;
  int i = gid >> 5, h = gid & 31;
  float di = dinv[i];
  float v = out[gid] + xw[gid] * di * di + b[h];
  out[gid] = relu ? fmaxf(v, 0.f) : v;
}

// pooled sums + counts (mean pool pieces)
__global__ void k_pool(const float* __restrict__ h2,
                       const int* __restrict__ batch,
                       float* __restrict__ pool, float* __restrict__ cnt,
                       int N) {
  int gid  = blockIdx.x * blockDim.x + threadIdx.x;
  int i    = gid >> 3;
  if (i >= N) return;
  int part = gid & 7;
  int g = batch[i];
  float4 v = ((const float4*)(h2 + (long long)i * HCH))[part];
  float* o = pool + (long long)g * HCH + part * 4;
  atomicAdd(o + 0, v.x);
  atomicAdd(o + 1, v.y);
  atomicAdd(o + 2, v.z);
  atomicAdd(o + 3, v.w);
  if (part == 0) atomicAdd(&cnt[g], 1.0f);
}

// hcat[g, 0:64] = [pool_l[g]/max(cnt_l,1), pool_r[g]/max(cnt_r,1)]
__global__ void k_hcat(const float* __restrict__ pl, const float* __restrict__ cl,
                       const float* __restrict__ pr, const float* __restrict__ cr,
                       float* __restrict__ hcat, int G) {
  int gid = blockIdx.x * blockDim.x + threadIdx.x;
  if (gid >= G * 2 * HCH) return;
  int g = gid >> 6, c = gid & 63;
  float v = (c < HCH) ? pl[g * HCH + c] / fmaxf(cl[g], 1.f)
                      : pr[g * HCH + (c - HCH)] / fmaxf(cr[g], 1.f);
  hcat[gid] = v;
}

// out[g] = fb2 + sum_h relu(z[g,h]+fb1[h]) * fw2[h]
__global__ void k_head(const float* __restrict__ z,
                       const float* __restrict__ fb1,
                       const float* __restrict__ fw2,
                       const float* __restrict__ fb2,
                       float* __restrict__ out, int G) {
  int g = blockIdx.x * blockDim.x + threadIdx.x;
  if (g >= G) return;
  float acc = fb2[0];
#pragma unroll
  for (int h = 0; h < HCH; ++h)
    acc += fmaxf(z[(long long)g * HCH + h] + fb1[h], 0.f) * fw2[h];
  out[g] = acc;
}

// ------------------------------- launcher ----------------------------------

static inline int cdiv(long long a, int b) { return (int)((a + b - 1) / b); }

extern "C" void kernel_launch(void* const* d_in, const int* in_sizes, int n_in,
                              void* d_out, int out_size, void* d_ws, size_t ws_size,
                              hipStream_t stream) {
  const int N = in_sizes[0];        // 100000 (multiple of 32)
  const int E = in_sizes[1] / 2;    // 2500000
  const int G = out_size;           // 1024  (multiple of 32)

  const int*   lhs_x  = (const int*)d_in[0];
  const int*   lhs_ei = (const int*)d_in[1];
  const int*   lhs_b  = (const int*)d_in[2];
  const int*   rhs_x  = (const int*)d_in[3];
  const int*   rhs_ei = (const int*)d_in[4];
  const int*   rhs_b  = (const int*)d_in[5];
  const float* emb    = (const float*)d_in[6];
  const float* W1     = (const float*)d_in[7];
  const float* b1     = (const float*)d_in[8];
  const float* W2     = (const float*)d_in[9];
  const float* b2     = (const float*)d_in[10];
  const float* fw1    = (const float*)d_in[11];
  const float* fb1    = (const float*)d_in[12];
  const float* fw2    = (const float*)d_in[13];
  const float* fb2    = (const float*)d_in[14];
  float* out = (float*)d_out;

  // workspace carve-up (floats); total ~6.7M floats ~ 27 MB
  float* ws     = (float*)d_ws;
  float* bufA   = ws;                               // N*32
  float* bufB   = bufA + (size_t)N * HCH;           // N*32
  float* dinv   = bufB + (size_t)N * HCH;           // N (deg -> dinv in place)
  float* pool_l = dinv + N;                         // G*32
  float* pool_r = pool_l + (size_t)G * HCH;         // G*32
  float* cnt_l  = pool_r + (size_t)G * HCH;         // G
  float* cnt_r  = cnt_l + G;                        // G
  float* hcat   = cnt_r + G;                        // G*64
  float* zbuf   = hcat + (size_t)G * 2 * HCH;       // G*32

  const int TPB = 256;
  const int nodeWaves  = N / 16;                    // 6250, even
  const int graphWaves = G / 16;                    // 64, even

  // zero pool sums + counts (contiguous block)
  {
    int n = 2 * G * HCH + 2 * G;
    k_set<<<cdiv(n, TPB), TPB, 0, stream>>>(pool_l, 0.f, n);
  }

  struct Side { const int* tok; const int* ei; const int* bat;
                float* pool; float* cnt; };
  Side sides[2] = { { lhs_x, lhs_ei, lhs_b, pool_l, cnt_l },
                    { rhs_x, rhs_ei, rhs_b, pool_r, cnt_r } };

  for (int sdx = 0; sdx < 2; ++sdx) {
    const int* src = sides[sdx].ei;
    const int* dst = sides[sdx].ei + E;

    // deg = 1 (self loop) + indegree; dinv = rsqrt(deg)
    k_set<<<cdiv(N, TPB), TPB, 0, stream>>>(dinv, 1.f, N);
    k_degree<<<cdiv(E, TPB), TPB, 0, stream>>>(dst, dinv, E);
    k_rsqrt<<<cdiv(N, TPB), TPB, 0, stream>>>(dinv, N);

    // conv1: bufA = emb[tok] @ W1 ; bufB = aggregate + self + b1
    k_embed_gemm<<<nodeWaves / 2, 64, 0, stream>>>(sides[sdx].tok, emb, W1, bufA);
    k_set<<<cdiv((long long)N * HCH, TPB), TPB, 0, stream>>>(bufB, 0.f, N * HCH);
    k_scatter<<<cdiv((long long)E * 8, TPB), TPB, 0, stream>>>(bufA, src, dst, dinv, bufB, E);
    k_selfbias<<<cdiv((long long)N * HCH, TPB), TPB, 0, stream>>>(bufA, dinv, b1, bufB, N, 0);

    // conv2: bufA = bufB @ W2 ; bufB = relu(aggregate + self + b2)
    k_gemm32<<<nodeWaves / 2, 64, 0, stream>>>(bufB, W2, bufA, HCH);
    k_set<<<cdiv((long long)N * HCH, TPB), TPB, 0, stream>>>(bufB, 0.f, N * HCH);
    k_scatter<<<cdiv((long long)E * 8, TPB), TPB, 0, stream>>>(bufA, src, dst, dinv, bufB, E);
    k_selfbias<<<cdiv((long long)N * HCH, TPB), TPB, 0, stream>>>(bufA, dinv, b2, bufB, N, 1);

    // mean-pool pieces
    k_pool<<<cdiv((long long)N * 8, TPB), TPB, 0, stream>>>(bufB, sides[sdx].bat,
                                                            sides[sdx].pool,
                                                            sides[sdx].cnt, N);
  }

  // head: hcat -> WMMA GEMM (K=64) -> bias/relu/dot
  k_hcat<<<cdiv((long long)G * 2 * HCH, TPB), TPB, 0, stream>>>(pool_l, cnt_l,
                                                                pool_r, cnt_r,
                                                                hcat, G);
  k_gemm32<<<graphWaves / 2, 64, 0, stream>>>(hcat, fw1, zbuf, 2 * HCH);
  k_head<<<cdiv(G, TPB), TPB, 0, stream>>>(zbuf, fb1, fw2, fb2, out, G);
}